// EdgePooling_56951266345245
// MI455X (gfx1250) — compile-verified
//
#include <hip/hip_runtime.h>
#include <hip/hip_bf16.h>
#include <math.h>

// ---------------- problem constants ----------------
#define N_GRAPHS   64
#define NODES_PER_G 1024
#define N_NODES    (N_GRAPHS * NODES_PER_G)
#define E_PER_G    4096
#define E_TOTAL    (N_GRAPHS * E_PER_G)
#define IN_CH      256
#define EDGE_CH    128
#define D_FEAT     (2 * IN_CH + EDGE_CH)          // 640
#define K_SEL      3277                            // ceil(0.8 * 4096)
#define R_TOTAL    (N_GRAPHS * K_SEL)              // 209728 surviving edges

// d_out layout (floats), concatenated in reference return order:
//  [0, 2*R_TOTAL)                     edge_index_out (src row then dst row)
//  [2*R_TOTAL, 2*R_TOTAL+R_TOTAL*128) edge_attr_out
//  [.., +R_TOTAL)                     edge_score_out
#define OUT_ATTR_OFF  (2 * R_TOTAL)
#define OUT_SCORE_OFF (2 * R_TOTAL + R_TOTAL * EDGE_CH)

typedef __attribute__((ext_vector_type(2))) float v2f;
typedef __attribute__((ext_vector_type(8))) float v8f;

// =====================================================================
// Kernel 1: edge scoring via V_WMMA_F32_16X16X4_F32.
// One wave scores 16 edges. A (16x4 f32): lane L (0-15) holds row M=L,
// K={0,1}; lane L+16 holds row M=L, K={2,3}  -> float2 per lane.
// B (4x16): column N = lane&15; N=0 -> Wf, N=1 -> Ws, N>=2 -> Wf (ignored).
// 160 WMMA steps accumulate both 640-wide dot products into C columns 0/1.
// =====================================================================
__global__ __launch_bounds__(256) void edgepool_score_kernel(
    const float* __restrict__ x,          // [N_NODES, 256]
    const int*   __restrict__ src,        // [E_TOTAL]
    const int*   __restrict__ dst,        // [E_TOTAL]
    const float* __restrict__ edge_attr,  // [E_TOTAL, 128]
    const float* __restrict__ Wf,         // [640]
    const float* __restrict__ bf,         // [1]
    const float* __restrict__ Ws,         // [640]
    const float* __restrict__ bs,         // [1]
    float*       __restrict__ scores)     // [E_TOTAL]
{
    __shared__ float sF[8][16];
    __shared__ float sS[8][16];

    const int wave  = threadIdx.x >> 5;
    const int lane  = threadIdx.x & 31;
    const int row   = lane & 15;          // M (edge within group) and N (col)
    const int khalf = lane >> 4;          // 0 -> K{0,1}, 1 -> K{2,3}

    const int egroup = blockIdx.x * 8 + wave;   // group of 16 edges
    const int e      = egroup * 16 + row;

    const int s  = src[e];
    const int dd = dst[e];
    const float* fxs = x + (size_t)s  * IN_CH  + khalf * 2;
    const float* fxd = x + (size_t)dd * IN_CH  + khalf * 2;
    const float* fea = edge_attr + (size_t)e * EDGE_CH + khalf * 2;
    // lanes with N>=2 read Wf too: harmless garbage columns, broadcast loads
    const float* w   = (row == 1) ? Ws : Wf;
    const float* wk  = w + khalf * 2;

    v8f c = {};

    // f in [0,256): x[src]
    #pragma unroll 4
    for (int kk = 0; kk < 64; ++kk) {
        v2f a = *(const v2f*)(fxs + kk * 4);
        v2f b = *(const v2f*)(wk  + kk * 4);
        c = __builtin_amdgcn_wmma_f32_16x16x4_f32(false, a, false, b,
                                                  (short)0, c, false, false);
    }
    // f in [256,512): x[dst]
    #pragma unroll 4
    for (int kk = 0; kk < 64; ++kk) {
        v2f a = *(const v2f*)(fxd + kk * 4);
        v2f b = *(const v2f*)(wk + 256 + kk * 4);
        c = __builtin_amdgcn_wmma_f32_16x16x4_f32(false, a, false, b,
                                                  (short)0, c, false, false);
    }
    // f in [512,640): edge_attr
    #pragma unroll 4
    for (int kk = 0; kk < 32; ++kk) {
        v2f a = *(const v2f*)(fea + kk * 4);
        v2f b = *(const v2f*)(wk + 512 + kk * 4);
        c = __builtin_amdgcn_wmma_f32_16x16x4_f32(false, a, false, b,
                                                  (short)0, c, false, false);
    }

    // C layout: VGPR v, lane l -> element (M = v + 8*(l>>4), N = l&15).
    // Columns N=0 (Wf dot) and N=1 (Ws dot) live in lanes {0,16} and {1,17}.
    if (row < 2) {
        float* o  = (row == 0) ? sF[wave] : sS[wave];
        const int mb = khalf * 8;
        #pragma unroll
        for (int v = 0; v < 8; ++v) o[mb + v] = c[v];
    }
    __syncthreads();

    if (lane < 16) {
        const float zf = sF[wave][row] + bf[0];
        const float zs = sS[wave][row] + bs[0];
        const float sig = 1.0f / (1.0f + __expf(-zf));
        const float sp  = (zs > 20.0f) ? zs : log1pf(__expf(zs));
        scores[e] = sig * sp;
    }
}

// =====================================================================
// Kernel 2: per-graph top-k via bitonic sort of 4096 packed keys in LDS.
// key = (~monotone(score)) << 32 | idx  -> ascending u64 sort gives
// descending score with ascending-index tie break (matches lax.top_k).
// One 256-thread block per graph; 32 KB LDS.
// =====================================================================
__global__ __launch_bounds__(256) void edgepool_topk_kernel(
    const float* __restrict__ scores,   // [E_TOTAL]
    const int*   __restrict__ src,
    const int*   __restrict__ dst,
    int*         __restrict__ perm,     // [R_TOTAL] selected global edge ids
    float*       __restrict__ out)      // full d_out
{
    __shared__ unsigned long long keys[E_PER_G];

    const int g   = blockIdx.x;
    const int tid = threadIdx.x;
    const float* sc = scores + (size_t)g * E_PER_G;

    for (int i = tid; i < E_PER_G; i += 256) {
        unsigned u = __float_as_uint(sc[i]);
        unsigned mono = (u & 0x80000000u) ? ~u : (u | 0x80000000u);
        keys[i] = ((unsigned long long)(~mono) << 32) | (unsigned)i;
    }

    for (int size = 2; size <= E_PER_G; size <<= 1) {
        for (int stride = size >> 1; stride > 0; stride >>= 1) {
            __syncthreads();
            for (int t = tid; t < E_PER_G / 2; t += 256) {
                const int i = ((t & ~(stride - 1)) << 1) | (t & (stride - 1));
                const int j = i + stride;
                const bool asc = ((i & size) == 0);
                const unsigned long long a = keys[i];
                const unsigned long long b = keys[j];
                if ((a > b) == asc) { keys[i] = b; keys[j] = a; }
            }
        }
    }
    __syncthreads();

    for (int t = tid; t < K_SEL; t += 256) {
        const int idx = (int)(keys[t] & 0xFFFFFFFFu);
        const int gid = g * E_PER_G + idx;
        const int o   = g * K_SEL + t;
        perm[o]                 = gid;
        out[o]                  = (float)src[gid];      // edge_index row 0
        out[R_TOTAL + o]        = (float)dst[gid];      // edge_index row 1
        out[OUT_SCORE_OFF + o]  = sc[idx];              // edge_score_out
    }
}

// =====================================================================
// Kernel 3: gather surviving edge_attr rows. One wave per output row,
// 32 lanes x float4 = 128 floats -> b128 coalesced read + write.
// =====================================================================
__global__ __launch_bounds__(256) void edgepool_attr_kernel(
    const float* __restrict__ edge_attr,  // [E_TOTAL, 128]
    const int*   __restrict__ perm,       // [R_TOTAL]
    float*       __restrict__ out)        // full d_out
{
    const int wave = threadIdx.x >> 5;
    const int lane = threadIdx.x & 31;
    const int r = blockIdx.x * 8 + wave;
    if (r >= R_TOTAL) return;
    const int gid = perm[r];
    const float4* s = (const float4*)(edge_attr + (size_t)gid * EDGE_CH);
    float4*       d = (float4*)(out + OUT_ATTR_OFF + (size_t)r * EDGE_CH);
    d[lane] = s[lane];
}

// =====================================================================
extern "C" void kernel_launch(void* const* d_in, const int* in_sizes, int n_in,
                              void* d_out, int out_size, void* d_ws, size_t ws_size,
                              hipStream_t stream)
{
    const float* x         = (const float*)d_in[0];
    const int*   edge_idx  = (const int*)d_in[1];     // [2, E_TOTAL]
    const float* edge_attr = (const float*)d_in[2];
    // d_in[3] = batch (unused)
    const float* Wf        = (const float*)d_in[4];
    const float* bf        = (const float*)d_in[5];
    const float* Ws        = (const float*)d_in[6];
    const float* bs        = (const float*)d_in[7];
    // d_in[8] = num_graphs (compile-time constant here)

    const int* src = edge_idx;
    const int* dst = edge_idx + E_TOTAL;

    float* scores = (float*)d_ws;                                   // 1 MB
    int*   perm   = (int*)((char*)d_ws + (size_t)E_TOTAL * sizeof(float));

    float* out = (float*)d_out;

    // 1) score all edges: 16 edges/wave, 8 waves/block -> 128 edges/block
    edgepool_score_kernel<<<E_TOTAL / 128, 256, 0, stream>>>(
        x, src, dst, edge_attr, Wf, bf, Ws, bs, scores);

    // 2) per-graph top-k + index/score outputs
    edgepool_topk_kernel<<<N_GRAPHS, 256, 0, stream>>>(
        scores, src, dst, perm, out);

    // 3) attr gather: 8 rows per block
    edgepool_attr_kernel<<<(R_TOTAL + 7) / 8, 256, 0, stream>>>(
        edge_attr, perm, out);
}